// _ResBlock_36885179138564
// MI455X (gfx1250) — compile-verified
//
#include <hip/hip_runtime.h>
#include <hip/hip_bf16.h>
#include <stdint.h>

typedef __attribute__((ext_vector_type(2))) float v2f;
typedef __attribute__((ext_vector_type(8))) float v8f;
typedef __attribute__((ext_vector_type(4))) unsigned int v4u;
typedef __attribute__((ext_vector_type(8))) int v8i;
typedef __attribute__((ext_vector_type(4))) int v4i;

#define C 256
#define LDS_STRIDE 260      // 256 + 4 floats: rows offset by 4 banks -> conflict-free A frags
#define M_TILES 2           // 32 nodes per block: halves weight L2 traffic vs 16
#define ROWS (16 * M_TILES)

#if defined(__has_builtin)
#if __has_builtin(__builtin_amdgcn_tensor_load_to_lds) && __has_builtin(__builtin_amdgcn_s_wait_tensorcnt)
#define USE_TDM 1
#endif
#endif
#ifndef USE_TDM
#define USE_TDM 0
#endif

#if USE_TDM
#warning "CDNA5: TDM path ACTIVE (tensor_load_to_lds + s_wait_tensorcnt)"
#else
#warning "CDNA5: TDM builtins UNAVAILABLE - using plain staging loads"
#endif

// ---------------------------------------------------------------------------
// Kernel 1: in-degree count. Edge list is streamed once -> NT loads keep the
// 51MB index stream from evicting x/agg (both L2-resident) from the 192MB L2.
// ---------------------------------------------------------------------------
__global__ __launch_bounds__(256) void _sage_count_kernel(
    const long long* __restrict__ ei, float* __restrict__ cnt, int n_edges) {
    int e = blockIdx.x * blockDim.x + threadIdx.x;
    if (e < n_edges) {
        int d = (int)__builtin_nontemporal_load(&ei[(size_t)n_edges + e]);
        unsafeAtomicAdd(&cnt[d], 1.0f);          // native global_atomic_add_f32
    }
}

// ---------------------------------------------------------------------------
// Kernel 2: scatter-add of x[src] into agg[dst]. 256 threads = one channel
// each; 64 edges staged in LDS per block. Coalesced 1KB gather + 1KB-wide
// f32 atomic per edge, all L2-resident (x, agg = 100 MB each vs 192 MB L2).
// ---------------------------------------------------------------------------
#define EDGES_PER_BLOCK 64
__global__ __launch_bounds__(256) void _sage_scatter_kernel(
    const float* __restrict__ x, const long long* __restrict__ ei,
    float* __restrict__ agg, int n_edges) {
    __shared__ int s_src[EDGES_PER_BLOCK];
    __shared__ int s_dst[EDGES_PER_BLOCK];
    int tid = threadIdx.x;
    int e0  = blockIdx.x * EDGES_PER_BLOCK;
    if (tid < EDGES_PER_BLOCK) {
        int e = e0 + tid;
        if (e < n_edges) {
            s_src[tid] = (int)__builtin_nontemporal_load(&ei[e]);
            s_dst[tid] = (int)__builtin_nontemporal_load(&ei[(size_t)n_edges + e]);
        }
    }
    __syncthreads();
    int nloc = n_edges - e0;
    if (nloc > EDGES_PER_BLOCK) nloc = EDGES_PER_BLOCK;
    for (int i = 0; i < nloc; ++i) {
        if (i + 2 < nloc) {  // cover gather latency: global_prefetch_b8
            __builtin_prefetch(&x[(size_t)s_src[i + 2] * C + tid], 0, 1);
        }
        int s = s_src[i];
        int d = s_dst[i];
        float v = x[(size_t)s * C + tid];
        unsafeAtomicAdd(&agg[(size_t)d * C + tid], v);
    }
}

// ---------------------------------------------------------------------------
// TDM: async-DMA a [tile_rows x 256] f32 tile (row stride 256) from global
// into LDS with 4-DWORD padding after each 1024B row -> LDS stride 260 floats.
// D# layout per CDNA5 ISA section 8 (group0 + group1; groups 2/3 zero = 2D).
// ---------------------------------------------------------------------------
#if USE_TDM
__device__ __forceinline__ void tdm_load_tile(unsigned lds_addr,
                                              const float* gptr,
                                              unsigned tensor_rows,
                                              unsigned tile_rows) {
    unsigned long long ga = (unsigned long long)(uintptr_t)gptr;
    v4u g0;
    g0[0] = 1u;                                        // count=1 (valid), no gather
    g0[1] = lds_addr;                                  // lds_addr [63:32]
    g0[2] = (unsigned)(ga & 0xFFFFFFFFull);            // global_addr [95:64]
    g0[3] = (unsigned)((ga >> 32) & 0x01FFFFFFull)     // global_addr [120:96]
            | 0x80000000u;                             // type=2 ("image") [127:126]
    v8i g1;
    g1[0] = (2 << 16)        // data_size = 4 bytes
          | (1 << 20)        // pad_enable
          | (7 << 22)        // pad_interval: 256 DWORDs (=1024B) per pad
          | (3 << 25);       // pad_amount: 4 DWORDs (=16B) -> stride 260 floats
    g1[1] = (int)(256u << 16);                         // tensor_dim0[15:0]=256
    g1[2] = (int)((tensor_rows & 0xFFFFu) << 16);      // tensor_dim1[15:0]
    g1[3] = (int)(((tensor_rows >> 16) & 0xFFFFu)      // tensor_dim1[31:16]
            | (256u << 16));                           // tile_dim0 = 256
    g1[4] = (int)tile_rows;                            // tile_dim1
    g1[5] = 256;                                       // tensor_dim0_stride
    g1[6] = 0;
    g1[7] = 0;
    v4i gz = {0, 0, 0, 0};
#if defined(__clang_major__) && (__clang_major__ >= 23)
    v8i gz8 = {0, 0, 0, 0, 0, 0, 0, 0};
    __builtin_amdgcn_tensor_load_to_lds(g0, g1, gz, gz, gz8, 0);
#else
    __builtin_amdgcn_tensor_load_to_lds(g0, g1, gz, gz, 0);
#endif
}
#endif

// ---------------------------------------------------------------------------
// Kernel 3: fused  h = (agg/cnt)@Wl^T + b_l + x@Wr^T ; LayerNorm ; x+relu(h)
// Block = 32 nodes x 256 ch, 8 waves; wave w owns N-tiles 32w and 32w+16 and
// both M-tiles => each B fragment feeds 2 WMMAs (halves weight L2 traffic).
// K loop: V_WMMA_F32_16X16X4_F32 (exact f32), A frags from TDM-staged LDS.
// ---------------------------------------------------------------------------
__global__ __launch_bounds__(256) void _sage_fused_kernel(
    const float* __restrict__ x, const float* __restrict__ agg,
    const float* __restrict__ cnt,
    const float* __restrict__ Wl, const float* __restrict__ bl,
    const float* __restrict__ Wr,
    const float* __restrict__ gamma, const float* __restrict__ beta,
    float* __restrict__ out, int n_nodes) {
    __shared__ __align__(16) float sA[ROWS * LDS_STRIDE];  // agg tile -> later h
    __shared__ __align__(16) float sX[ROWS * LDS_STRIDE];  // x tile (residual)
    __shared__ float sMu[ROWS];
    __shared__ float sRs[ROWS];

    const int tid   = threadIdx.x;
    const int wave  = tid >> 5;
    const int lane  = tid & 31;
    const int node0 = blockIdx.x * ROWS;

    // ---- stage agg & x tiles into LDS ----
#if USE_TDM
    if (wave == 0) {   // TDM ignores EXEC; issue once from wave 0 (uniform ops)
        tdm_load_tile((unsigned)(uintptr_t)sA, agg + (size_t)node0 * C,
                      (unsigned)n_nodes, ROWS);
        tdm_load_tile((unsigned)(uintptr_t)sX, x + (size_t)node0 * C,
                      (unsigned)n_nodes, ROWS);
        __builtin_amdgcn_s_wait_tensorcnt(0);
    }
    __syncthreads();
    // scale agg rows by 1/max(cnt,1) in LDS
    for (int r = 0; r < ROWS; ++r) {
        int node = node0 + r;
        if (node >= n_nodes) break;
        float inv = 1.0f / fmaxf(cnt[node], 1.0f);
        sA[r * LDS_STRIDE + tid] *= inv;
    }
#else
    for (int r = 0; r < ROWS; ++r) {
        int node = node0 + r;
        if (node >= n_nodes) node = n_nodes - 1;
        float inv = 1.0f / fmaxf(cnt[node], 1.0f);
        sA[r * LDS_STRIDE + tid] = agg[(size_t)node * C + tid] * inv;
        sX[r * LDS_STRIDE + tid] = x[(size_t)node * C + tid];
    }
#endif
    __syncthreads();

    // f32 WMMA fragment mapping (ISA 7.12.2):
    //  A 16x4: lane L -> M=L%16, vgpr0=K(L<16?0:2), vgpr1=K+1
    //  B 4x16: lane L -> N=L%16, same K split
    //  C/D 16x16: vgpr r: lanes 0-15 (M=r), lanes 16-31 (M=r+8)
    const int m16   = lane & 15;
    const int khalf = (lane >> 4) << 1;
    const int n0    = wave * 32;
    const int n1    = n0 + 16;

    v8f acc00 = {}, acc01 = {}, acc10 = {}, acc11 = {};

    for (int k0 = 0; k0 < C; k0 += 4) {
        const int kb = k0 + khalf;
        v2f aA0 = *(const v2f*)&sA[m16 * LDS_STRIDE + kb];
        v2f aA1 = *(const v2f*)&sA[(16 + m16) * LDS_STRIDE + kb];
        v2f aX0 = *(const v2f*)&sX[m16 * LDS_STRIDE + kb];
        v2f aX1 = *(const v2f*)&sX[(16 + m16) * LDS_STRIDE + kb];
        v2f bL0 = *(const v2f*)&Wl[(size_t)(n0 + m16) * C + kb];
        v2f bL1 = *(const v2f*)&Wl[(size_t)(n1 + m16) * C + kb];
        v2f bR0 = *(const v2f*)&Wr[(size_t)(n0 + m16) * C + kb];
        v2f bR1 = *(const v2f*)&Wr[(size_t)(n1 + m16) * C + kb];

        acc00 = __builtin_amdgcn_wmma_f32_16x16x4_f32(false, aA0, false, bL0,
                                                      (short)0, acc00, false, false);
        acc00 = __builtin_amdgcn_wmma_f32_16x16x4_f32(false, aX0, false, bR0,
                                                      (short)0, acc00, false, false);
        acc01 = __builtin_amdgcn_wmma_f32_16x16x4_f32(false, aA0, false, bL1,
                                                      (short)0, acc01, false, false);
        acc01 = __builtin_amdgcn_wmma_f32_16x16x4_f32(false, aX0, false, bR1,
                                                      (short)0, acc01, false, false);
        acc10 = __builtin_amdgcn_wmma_f32_16x16x4_f32(false, aA1, false, bL0,
                                                      (short)0, acc10, false, false);
        acc10 = __builtin_amdgcn_wmma_f32_16x16x4_f32(false, aX1, false, bR0,
                                                      (short)0, acc10, false, false);
        acc11 = __builtin_amdgcn_wmma_f32_16x16x4_f32(false, aA1, false, bL1,
                                                      (short)0, acc11, false, false);
        acc11 = __builtin_amdgcn_wmma_f32_16x16x4_f32(false, aX1, false, bR1,
                                                      (short)0, acc11, false, false);
    }
    __syncthreads();                 // done reading sA -> reuse as h

    float* sH = sA;
    {
        const float bln0 = bl[n0 + m16];
        const float bln1 = bl[n1 + m16];
        const int rbase = (lane >> 4) << 3;   // 0 or 8
        for (int r = 0; r < 8; ++r) {
            sH[(rbase + r) * LDS_STRIDE + n0 + m16]      = acc00[r] + bln0;
            sH[(rbase + r) * LDS_STRIDE + n1 + m16]      = acc01[r] + bln1;
            sH[(16 + rbase + r) * LDS_STRIDE + n0 + m16] = acc10[r] + bln0;
            sH[(16 + rbase + r) * LDS_STRIDE + n1 + m16] = acc11[r] + bln1;
        }
    }
    __syncthreads();

    // ---- LayerNorm stats: 16 threads/row, shfl_xor tree inside 16-lane half
    for (int half = 0; half < M_TILES; ++half) {
        const int r = (tid >> 4) + 16 * half;
        const int j = tid & 15;
        float s = 0.0f, ss = 0.0f;
        for (int i = 0; i < 16; ++i) {
            float v = sH[r * LDS_STRIDE + j + 16 * i];
            s  += v;
            ss += v * v;
        }
        for (int off = 8; off > 0; off >>= 1) {
            s  += __shfl_xor(s,  off, 32);
            ss += __shfl_xor(ss, off, 32);
        }
        if (j == 0) {
            float mu  = s * (1.0f / 256.0f);
            float var = ss * (1.0f / 256.0f) - mu * mu;
            sMu[r] = mu;
            sRs[r] = rsqrtf(var + 1e-5f);
        }
    }
    __syncthreads();

    // ---- epilogue: LN affine + residual + relu. Output is written once and
    // never re-read -> nontemporal stores preserve L2 residency of x/agg/W.
    const float g  = gamma[tid];
    const float bt = beta[tid];
    for (int r = 0; r < ROWS; ++r) {
        int node = node0 + r;
        if (node >= n_nodes) break;
        float hv  = sH[r * LDS_STRIDE + tid];
        float val = (hv - sMu[r]) * sRs[r] * g + bt;
        float xv  = sX[r * LDS_STRIDE + tid];
        __builtin_nontemporal_store(xv + fmaxf(val, 0.0f),
                                    &out[(size_t)node * C + tid]);
    }
}

// ---------------------------------------------------------------------------
extern "C" void kernel_launch(void* const* d_in, const int* in_sizes, int n_in,
                              void* d_out, int out_size, void* d_ws, size_t ws_size,
                              hipStream_t stream) {
    const float*     x     = (const float*)d_in[0];
    const long long* ei    = (const long long*)d_in[1];   // int64 [2, E]
    const float*     Wl    = (const float*)d_in[2];
    const float*     bl    = (const float*)d_in[3];
    const float*     Wr    = (const float*)d_in[4];
    const float*     gamma = (const float*)d_in[5];
    const float*     beta  = (const float*)d_in[6];
    float*           out   = (float*)d_out;

    const int n_nodes = in_sizes[0] / C;
    const int n_edges = in_sizes[1] / 2;

    float* agg = (float*)d_ws;
    float* cnt = agg + (size_t)n_nodes * C;

    hipMemsetAsync(d_ws, 0,
                   ((size_t)n_nodes * C + (size_t)n_nodes) * sizeof(float),
                   stream);

    int cblocks = (n_edges + 255) / 256;
    _sage_count_kernel<<<cblocks, 256, 0, stream>>>(ei, cnt, n_edges);

    int sblocks = (n_edges + EDGES_PER_BLOCK - 1) / EDGES_PER_BLOCK;
    _sage_scatter_kernel<<<sblocks, 256, 0, stream>>>(x, ei, agg, n_edges);

    int fblocks = (n_nodes + ROWS - 1) / ROWS;
    _sage_fused_kernel<<<fblocks, 256, 0, stream>>>(
        x, agg, cnt, Wl, bl, Wr, gamma, beta, out, n_nodes);
}